// CombSubFast_32109175505144
// MI455X (gfx1250) — compile-verified
//
#include <hip/hip_runtime.h>
#include <math.h>

// ---------------- problem constants ----------------
#define B_    8
#define NFR   1024                 // input frames
#define BLK   512
#define T_    (NFR*BLK)            // 524288 samples per batch
#define NFFT  1024
#define NBINS 513
#define NBPAD 640                  // forward table column padding (5 * 128)
#define MFR   (NFR+1)              // 1025 frames per batch after extend
#define MTOT  (B_*MFR)             // 8200 frames total
#define MPAD  8208                 // 513 * 16
#define MTILES 513
#define SIGC  528                  // stored spectrum columns (33*16)
#define KINV  528
#define SRF   44100.0f
#define PI_F  3.14159265358979323846f

#define CHUNK 4096
#define CTH   256
#define PERTH 16
#define NCHUNK (T_/CHUNK)          // 128

typedef float v2f __attribute__((ext_vector_type(2)));
typedef float v4f __attribute__((ext_vector_type(4)));
typedef float v8f __attribute__((ext_vector_type(8)));

// ---------------- helpers ----------------
__device__ __forceinline__ float f0_at(const float* __restrict__ f0f, int b, int t) {
    int i0 = t >> 9;                       // / BLK
    float frac = (float)(t & (BLK-1)) * (1.0f/(float)BLK);
    int i1 = i0 + 1; if (i1 > NFR-1) i1 = NFR-1;
    float a = f0f[b*NFR + i0];
    float c = f0f[b*NFR + i1];
    return a * (1.0f - frac) + c * frac;
}

// ---------------- K0: DFT basis tables, pre-swizzled into WMMA-B order ----
// fwdP[kpair][bin][4] = { win(2p)*cos(2p,f), win(2p+1)*cos(2p+1,f),
//                        -win(2p)*sin(2p,f),-win(2p+1)*sin(2p+1,f) }
// => one 16B fragment load feeds both the cos- and sin-WMMA B operands.
__global__ void build_fwd_tables(float* __restrict__ fwdP) {
    int idx = blockIdx.x * blockDim.x + threadIdx.x;
    if (idx >= (NFFT/2)*NBPAD) return;
    int p = idx / NBPAD, f = idx % NBPAD;
    float out[4];
    for (int j = 0; j < 2; ++j) {
        int k = 2*p + j;
        float win = sinf(PI_F * (float)k / (float)NFFT);   // == sqrt-hann window
        int kf = (k * f) & (NFFT - 1);
        float ang = 2.0f * PI_F * (float)kf / (float)NFFT;
        out[j]     =  win * cosf(ang);
        out[2 + j] = -win * sinf(ang);
    }
    v4f v = { out[0], out[1], out[2], out[3] };
    *(v4f*)(fwdP + (size_t)idx * 4) = v;
}

// invP[fpair][t][4] = { invC(2p,t), invC(2p+1,t), invS(2p,t), invS(2p+1,t) }
// with window, 1/N and Hermitian x2 factors folded in; rows f>=513 zero.
__global__ void build_inv_tables(float* __restrict__ invP) {
    int idx = blockIdx.x * blockDim.x + threadIdx.x;
    if (idx >= (KINV/2)*NFFT) return;
    int p = idx / NFFT, t = idx % NFFT;
    float win = sinf(PI_F * (float)t / (float)NFFT);
    float out[4];
    for (int j = 0; j < 2; ++j) {
        int f = 2*p + j;
        if (f >= NBINS) { out[j] = 0.f; out[2 + j] = 0.f; continue; }
        float a = (f == 0 || f == NFFT/2) ? 1.0f : 2.0f;
        int ft = (f * t) & (NFFT - 1);
        float ang = 2.0f * PI_F * (float)ft / (float)NFFT;
        float s = win * a * (1.0f/(float)NFFT);
        out[j]     =  s * cosf(ang);
        out[2 + j] = -s * sinf(ang);
    }
    v4f v = { out[0], out[1], out[2], out[3] };
    *(v4f*)(invP + (size_t)idx * 4) = v;
}

// ---------------- K1: per-chunk partial sums of f0/sr ----------------
__global__ __launch_bounds__(CTH) void scan_partial(const float* __restrict__ f0f,
                                                    float* __restrict__ partial) {
    __shared__ float red[CTH];
    int b = blockIdx.y, c = blockIdx.x, tid = threadIdx.x;
    int t0 = c*CHUNK + tid*PERTH;
    float s = 0.f;
    for (int i = 0; i < PERTH; ++i) s += f0_at(f0f, b, t0 + i) * (1.0f/SRF);
    red[tid] = s; __syncthreads();
    for (int o = CTH/2; o > 0; o >>= 1) {
        if (tid < o) red[tid] += red[tid + o];
        __syncthreads();
    }
    if (tid == 0) partial[b*NCHUNK + c] = red[0];
}

// ---------------- K2: exclusive scan over chunk sums (one block / batch) ----
__global__ void scan_chunks(const float* __restrict__ partial, float* __restrict__ prefix) {
    __shared__ float sh[NCHUNK];
    int b = blockIdx.x, tid = threadIdx.x;
    float v = partial[b*NCHUNK + tid];
    sh[tid] = v; __syncthreads();
    for (int o = 1; o < NCHUNK; o <<= 1) {
        float add = (tid >= o) ? sh[tid - o] : 0.f;
        __syncthreads();
        sh[tid] += add;
        __syncthreads();
    }
    prefix[b*NCHUNK + tid] = sh[tid] - v;   // exclusive
}

// ---------------- K3: comb signal: x=cumsum(f0/sr), sinc ----------------
__global__ __launch_bounds__(CTH) void comb_k(const float* __restrict__ f0f,
                                              const float* __restrict__ prefix,
                                              float* __restrict__ comb) {
    __shared__ float sh[CTH];
    int b = blockIdx.y, c = blockIdx.x, tid = threadIdx.x;
    int t0 = c*CHUNK + tid*PERTH;
    float f0r[PERTH];
    float s = 0.f;
    for (int i = 0; i < PERTH; ++i) {
        float v = f0_at(f0f, b, t0 + i);
        f0r[i] = v;
        s += v * (1.0f/SRF);
    }
    sh[tid] = s; __syncthreads();
    for (int o = 1; o < CTH; o <<= 1) {
        float add = (tid >= o) ? sh[tid - o] : 0.f;
        __syncthreads();
        sh[tid] += add;
        __syncthreads();
    }
    float x = prefix[b*NCHUNK + c] + (sh[tid] - s);   // exclusive within chunk
    for (int i = 0; i < PERTH; ++i) {
        x += f0r[i] * (1.0f/SRF);
        float xr = x - rintf(x);
        float z = SRF * xr / (f0r[i] + 0.001f);
        float v = (fabsf(z) < 1e-6f) ? 1.0f : (sinf(PI_F*z) / (PI_F*z));
        comb[(size_t)b*T_ + t0 + i] = v;
    }
}

// ---------------- K4: forward rDFT (WMMA f32) + spectral filter ------------
// block: 256 thr = 8 waves; tile: 16 frames x 128 bins (16 bins per wave)
__global__ __launch_bounds__(256) void fwd_gemm(
    const float* __restrict__ comb, const float* __restrict__ noiseU,
    const float* __restrict__ hmag, const float* __restrict__ hph,
    const float* __restrict__ nmag,
    const float* __restrict__ fwdP,
    float* __restrict__ sigRe, float* __restrict__ sigIm)
{
    __shared__ float Act[16][260];   // +4 pad: conflict-free b64 LDS reads
    __shared__ float Anz[16][260];
    const int tid  = threadIdx.x;
    const int wave = tid >> 5, lane = tid & 31;
    const int half = lane >> 4, ln = lane & 15;
    const int mBase = blockIdx.y * 16;
    const int bin   = blockIdx.x * 128 + wave * 16 + ln;   // N column, < 640

    v8f accCtRe = {}; v8f accCtIm = {}; v8f accNzRe = {}; v8f accNzIm = {};

    for (int kc = 0; kc < NFFT; kc += 256) {
        __syncthreads();
        // cooperative frame gather (window lives in the B tables)
        for (int i = tid; i < 16*256; i += 256) {
            int r = i >> 8, c = i & 255;
            int m = mBase + r;
            float vc = 0.f, vn = 0.f;
            if (m < MTOT) {
                int b = m / MFR, j = m - b*MFR;
                int s = j*BLK + kc + c - BLK;              // zero-padded frame
                if (s >= 0 && s < T_) {
                    vc = comb  [(size_t)b*T_ + s];
                    vn = noiseU[(size_t)b*T_ + s] * 2.0f - 1.0f;
                }
            }
            Act[r][c] = vc; Anz[r][c] = vn;
        }
        __syncthreads();
        for (int kk = 0; kk < 64; ++kk) {
            int kl = kk*4 + half*2;                        // A/B K-pair per lane half
            v2f aCt = *(const v2f*)&Act[ln][kl];
            v2f aNz = *(const v2f*)&Anz[ln][kl];
            int kg = kc + kl;                              // even
            // one b128 = whole B fragment pair (cos & sin halves)
            v4f bPk = *(const v4f*)(fwdP + (((size_t)(kg >> 1))*NBPAD + bin) * 4);
            v2f bC = { bPk.x, bPk.y };
            v2f bS = { bPk.z, bPk.w };
            accCtRe = __builtin_amdgcn_wmma_f32_16x16x4_f32(false, aCt, false, bC, (short)0, accCtRe, false, false);
            accCtIm = __builtin_amdgcn_wmma_f32_16x16x4_f32(false, aCt, false, bS, (short)0, accCtIm, false, false);
            accNzRe = __builtin_amdgcn_wmma_f32_16x16x4_f32(false, aNz, false, bC, (short)0, accNzRe, false, false);
            accNzIm = __builtin_amdgcn_wmma_f32_16x16x4_f32(false, aNz, false, bS, (short)0, accNzIm, false, false);
        }
    }

    // fused filter epilogue: sig = ct_fft * exp(hm + i*pi*hp) + nz_fft * exp(nm)/128
    for (int i = 0; i < 8; ++i) {
        int m = mBase + half*8 + i;                        // D row: i + 8*(lane>=16)
        if (m >= MTOT || bin >= SIGC) continue;
        float sRe = 0.f, sIm = 0.f;
        if (bin < NBINS) {
            int b = m / MFR, j = m - b*MFR;
            int jj = (j < NFR) ? j : NFR-1;                // extend(): repeat last frame
            size_t fi = ((size_t)b*NFR + jj)*NBINS + bin;
            float eh  = expf(hmag[fi]);
            float ph  = PI_F * hph[fi];
            float hRe = eh * cosf(ph), hIm = eh * sinf(ph);
            float nf  = expf(nmag[fi]) * (1.0f/128.0f);
            float cr = accCtRe[i], ci = accCtIm[i];
            sRe = cr*hRe - ci*hIm + accNzRe[i]*nf;
            sIm = cr*hIm + ci*hRe + accNzIm[i]*nf;
        }
        sigRe[(size_t)m*SIGC + bin] = sRe;
        sigIm[(size_t)m*SIGC + bin] = sIm;
    }
}

// ---------------- K5: inverse rDFT (WMMA f32), windowed frames -------------
__global__ __launch_bounds__(256) void inv_gemm(
    const float* __restrict__ sigRe, const float* __restrict__ sigIm,
    const float* __restrict__ invP,
    float* __restrict__ fout)
{
    __shared__ float Are[16][268];
    __shared__ float Aim[16][268];
    const int tid  = threadIdx.x;
    const int wave = tid >> 5, lane = tid & 31;
    const int half = lane >> 4, ln = lane & 15;
    const int mBase = blockIdx.y * 16;
    const int tCol  = blockIdx.x * 128 + wave * 16 + ln;   // output sample in frame

    v8f acc = {};
    for (int fc = 0; fc < KINV; fc += 264) {
        __syncthreads();
        for (int i = tid; i < 16*264; i += 256) {
            int r = i / 264, c = i - r*264;
            int m = mBase + r;
            float vr = 0.f, vi = 0.f;
            if (m < MTOT) {
                size_t idx = (size_t)m*SIGC + fc + c;
                vr = sigRe[idx]; vi = sigIm[idx];
            }
            Are[r][c] = vr; Aim[r][c] = vi;
        }
        __syncthreads();
        for (int kk = 0; kk < 66; ++kk) {
            int kl = kk*4 + half*2;
            v2f aR = *(const v2f*)&Are[ln][kl];
            v2f aI = *(const v2f*)&Aim[ln][kl];
            int f = fc + kl;                               // even
            v4f bPk = *(const v4f*)(invP + (((size_t)(f >> 1))*NFFT + tCol) * 4);
            v2f bC = { bPk.x, bPk.y };
            v2f bS = { bPk.z, bPk.w };
            acc = __builtin_amdgcn_wmma_f32_16x16x4_f32(false, aR, false, bC, (short)0, acc, false, false);
            acc = __builtin_amdgcn_wmma_f32_16x16x4_f32(false, aI, false, bS, (short)0, acc, false, false);
        }
    }
    for (int i = 0; i < 8; ++i) {
        int m = mBase + half*8 + i;
        if (m < MTOT) fout[(size_t)m*NFFT + tCol] = acc[i];
    }
}

// ---------------- K6: deterministic overlap-add gather ----------------
__global__ void ola_k(const float* __restrict__ fout, float* __restrict__ out) {
    int idx = blockIdx.x * blockDim.x + threadIdx.x;
    if (idx >= B_*T_) return;
    int b = idx / T_, t = idx - b*T_;
    int pp = t + BLK;
    int q = pp >> 9, r = pp & (BLK-1);
    // out = h0 of frame q  +  h1 of frame q-1   (q in [1,1024], always valid)
    out[idx] = fout[((size_t)(b*MFR + q    ))*NFFT + r]
             + fout[((size_t)(b*MFR + q - 1))*NFFT + BLK + r];
}

// ---------------- launcher ----------------
extern "C" void kernel_launch(void* const* d_in, const int* in_sizes, int n_in,
                              void* d_out, int out_size, void* d_ws, size_t ws_size,
                              hipStream_t stream) {
    const float* f0f  = (const float*)d_in[0];
    const float* hmag = (const float*)d_in[1];
    const float* hph  = (const float*)d_in[2];
    const float* nmag = (const float*)d_in[3];
    const float* nzu  = (const float*)d_in[4];
    float* out = (float*)d_out;
    float* ws  = (float*)d_ws;

    size_t o = 0;
    float* comb  = ws + o; o += (size_t)B_*T_;             // 16 MB
    float* fwdP  = ws + o; o += (size_t)(NFFT/2)*NBPAD*4;  // 5.2 MB packed fwd
    float* invP  = ws + o; o += (size_t)(KINV/2)*NFFT*4;   // 4.3 MB packed inv
    float* sigRe = ws + o; o += (size_t)MPAD*SIGC;         // 17.3 MB
    float* sigIm = ws + o; o += (size_t)MPAD*SIGC;
    float* fout  = ws + o; o += (size_t)MPAD*NFFT;         // 33.6 MB
    float* part  = ws + o; o += (size_t)B_*NCHUNK;
    float* pref  = ws + o; o += (size_t)B_*NCHUNK;

    build_fwd_tables<<<((NFFT/2)*NBPAD + 255)/256, 256, 0, stream>>>(fwdP);
    build_inv_tables<<<((KINV/2)*NFFT + 255)/256, 256, 0, stream>>>(invP);
    scan_partial<<<dim3(NCHUNK, B_), CTH, 0, stream>>>(f0f, part);
    scan_chunks<<<B_, NCHUNK, 0, stream>>>(part, pref);
    comb_k<<<dim3(NCHUNK, B_), CTH, 0, stream>>>(f0f, pref, comb);
    fwd_gemm<<<dim3(5, MTILES), 256, 0, stream>>>(comb, nzu, hmag, hph, nmag,
                                                  fwdP, sigRe, sigIm);
    inv_gemm<<<dim3(8, MTILES), 256, 0, stream>>>(sigRe, sigIm, invP, fout);
    ola_k<<<(B_*T_ + 255)/256, 256, 0, stream>>>(fout, out);
}